// GINEEncoder_17463337025612
// MI455X (gfx1250) — compile-verified
//
#include <hip/hip_runtime.h>

#define N_NODES 50000
#define N_EDGES 600000
#define DIM 128
#define WPAD 132            // 132 dwords row stride -> conflict-free LDS banks
#define BN_EPS 1e-5f

typedef __attribute__((ext_vector_type(2))) float v2f;
typedef __attribute__((ext_vector_type(8))) float v8f;

// ---------- zero aggr + BN accumulators ----------
__global__ void k_zero(float* __restrict__ p, int n4) {
    int i = blockIdx.x * blockDim.x + threadIdx.x;
    if (i < n4) ((float4*)p)[i] = make_float4(0.f, 0.f, 0.f, 0.f);
}

// ---------- edge scatter: aggr[dst] += relu(x[src] + e) ----------
// one wave32 per edge; each lane owns a float4 (4 columns)
__global__ void k_scatter(const float* __restrict__ x,
                          const float* __restrict__ ea,
                          const int* __restrict__ src,
                          const int* __restrict__ dst,
                          float* __restrict__ aggr) {
    int t = blockIdx.x * blockDim.x + threadIdx.x;
    int e = t >> 5;
    int q = t & 31;
    if (e >= N_EDGES) return;
    int s = src[e];
    int d = dst[e];
    float4 xv = ((const float4*)(x  + (size_t)s * DIM))[q];
    float4 ev = ((const float4*)(ea + (size_t)e * DIM))[q];
    float4 m;
    m.x = fmaxf(xv.x + ev.x, 0.f);
    m.y = fmaxf(xv.y + ev.y, 0.f);
    m.z = fmaxf(xv.z + ev.z, 0.f);
    m.w = fmaxf(xv.w + ev.w, 0.f);
    float* o = aggr + (size_t)d * DIM + q * 4;
    atomicAdd(o + 0, m.x);
    atomicAdd(o + 1, m.y);
    atomicAdd(o + 2, m.z);
    atomicAdd(o + 3, m.w);
}

// ---------- h = relu((x+aggr) @ W^T + b), in-place over aggr; BN partials ----------
// block = 256 threads = 8 waves; block owns 16 rows, wave w owns cols [16w,16w+16)
__global__ void __launch_bounds__(256) k_gemm(
        const float* __restrict__ x, float* __restrict__ aggr,
        const float* __restrict__ W, const float* __restrict__ bias,
        float* __restrict__ colsum, float* __restrict__ colsumsq) {
    extern __shared__ float smem[];
    float* wl = smem;                 // [128][WPAD]  (W, row-major)
    float* zl = smem + DIM * WPAD;    // [16][WPAD]   (z = x + aggr tile)
    const int t  = threadIdx.x;
    const int r0 = blockIdx.x * 16;

    // stage W: 128x128 f32, coalesced float4 loads (L2 resident)
    #pragma unroll
    for (int i = 0; i < 16; ++i) {
        int c = t + i * 256;          // 0..4095 float4 chunks
        int row = c >> 5, kq = c & 31;
        float4 wv = ((const float4*)(W + (size_t)row * DIM))[kq];
        float* p = wl + row * WPAD + kq * 4;
        p[0] = wv.x; p[1] = wv.y; p[2] = wv.z; p[3] = wv.w;
    }
    // stage z tile = x + aggr
    #pragma unroll
    for (int i = 0; i < 2; ++i) {
        int c = t + i * 256;          // 0..511 float4 chunks
        int row = c >> 5, kq = c & 31;
        float4 xv = ((const float4*)(x    + (size_t)(r0 + row) * DIM))[kq];
        float4 av = ((const float4*)(aggr + (size_t)(r0 + row) * DIM))[kq];
        float* p = zl + row * WPAD + kq * 4;
        p[0] = xv.x + av.x; p[1] = xv.y + av.y;
        p[2] = xv.z + av.z; p[3] = xv.w + av.w;
    }
    __syncthreads();

    const int wave = t >> 5;          // column tile
    const int lane = t & 31;
    const int n    = lane & 15;       // output column within tile / B col
    const int m    = lane & 15;       // A row
    const int half = lane >> 4;

    // f32 WMMA fragment layout: K = vgpr_index + 2*lane_half
    const float* za = zl + m * WPAD + 2 * half;
    const float* wb = wl + (wave * 16 + n) * WPAD + 2 * half;   // B[k][n] = W[n_glob][k]

    v8f c = {};
    #pragma unroll 8
    for (int k0 = 0; k0 < DIM; k0 += 4) {
        v2f a = *(const v2f*)(za + k0);
        v2f b = *(const v2f*)(wb + k0);
        c = __builtin_amdgcn_wmma_f32_16x16x4_f32(
                false, a, false, b, (short)0, c, false, false);
    }

    const float bv = bias[wave * 16 + n];
    float s = 0.f, sq = 0.f;
    float h[8];
    #pragma unroll
    for (int r = 0; r < 8; ++r) {     // c[r] -> row (r0 + r + 8*half), col (16*wave + n)
        float v = fmaxf(c[r] + bv, 0.f);
        h[r] = v; s += v; sq += v * v;
    }
    float* o = aggr + (size_t)(r0 + 8 * half) * DIM + wave * 16 + n;
    #pragma unroll
    for (int r = 0; r < 8; ++r) o[(size_t)r * DIM] = h[r];

    // fold the two lane-halves (same column), then one atomic per column
    s  += __shfl_xor(s, 16, 32);
    sq += __shfl_xor(sq, 16, 32);
    if (half == 0) {
        atomicAdd(colsum   + wave * 16 + n, s);
        atomicAdd(colsumsq + wave * 16 + n, sq);
    }
}

// ---------- fold BN stats into scale/shift (in place) ----------
__global__ void k_bnfin(float* __restrict__ colsum, float* __restrict__ colsumsq,
                        const float* __restrict__ gamma, const float* __restrict__ beta) {
    int d = threadIdx.x;
    float mean  = colsum[d] * (1.f / N_NODES);
    float var   = colsumsq[d] * (1.f / N_NODES) - mean * mean;
    float scale = gamma[d] * rsqrtf(var + BN_EPS);
    colsum[d]   = scale;                    // -> scale
    colsumsq[d] = beta[d] - mean * scale;   // -> shift
}

// ---------- out = h*scale + shift ----------
__global__ void k_bnapply(const float* __restrict__ h,
                          const float* __restrict__ scale,
                          const float* __restrict__ shift,
                          float* __restrict__ out) {
    int i = blockIdx.x * blockDim.x + threadIdx.x;   // float4 index
    int q = i & 31;                                  // chunk within 128-col row
    float4 hv = ((const float4*)h)[i];
    float4 sc = ((const float4*)scale)[q];
    float4 sh = ((const float4*)shift)[q];
    float4 o;
    o.x = hv.x * sc.x + sh.x;
    o.y = hv.y * sc.y + sh.y;
    o.z = hv.z * sc.z + sh.z;
    o.w = hv.w * sc.w + sh.w;
    ((float4*)out)[i] = o;
}

extern "C" void kernel_launch(void* const* d_in, const int* in_sizes, int n_in,
                              void* d_out, int out_size, void* d_ws, size_t ws_size,
                              hipStream_t stream) {
    const int*   edge = (const int*)d_in[0];
    const int*   src  = edge;
    const int*   dst  = edge + N_EDGES;
    const float* node = (const float*)d_in[1];
    const float* ea   = (const float*)d_in[2];
    float* out = (float*)d_out;

    float* aggr     = (float*)d_ws;                      // N*D floats (aggr, then h in place)
    float* colsum   = aggr + (size_t)N_NODES * DIM;      // 128
    float* colsumsq = colsum + DIM;                      // 128

    const size_t shmem = (size_t)(DIM * WPAD + 16 * WPAD) * sizeof(float); // ~76 KB
    const int nzero4 = (N_NODES * DIM + 2 * DIM) / 4;

    for (int layer = 0; layer < 3; ++layer) {
        const float* W  = (const float*)d_in[3 + 4 * layer];
        const float* b  = (const float*)d_in[4 + 4 * layer];
        const float* g  = (const float*)d_in[5 + 4 * layer];
        const float* be = (const float*)d_in[6 + 4 * layer];
        const float* xin = (layer == 0) ? node : out;

        k_zero   <<<(nzero4 + 255) / 256, 256, 0, stream>>>(aggr, nzero4);
        k_scatter<<<(N_EDGES * 32) / 256, 256, 0, stream>>>(xin, ea, src, dst, aggr);
        k_gemm   <<<N_NODES / 16, 256, shmem, stream>>>(xin, aggr, W, b, colsum, colsumsq);
        k_bnfin  <<<1, DIM, 0, stream>>>(colsum, colsumsq, g, be);
        k_bnapply<<<(N_NODES * DIM / 4) / 256, 256, 0, stream>>>(aggr, colsum, colsumsq, out);
    }
}